// HiddenMarkovModel_42167988912676
// MI455X (gfx1250) — compile-verified
//
#include <hip/hip_runtime.h>
#include <stdint.h>

// HMM scaled-forward on gfx1250 (MI455X).
// Emissions: [x^2, x](16x128) @ W(128x32) via V_WMMA_F32_16X16X4_F32.
// Recurrence: alpha(16x32) @ P(32x32) per step via the same WMMA, with per-step
// rescaling; running log-scale == logsumexp(log_alpha[t]) == output.
// X chunks staged to LDS with global_load_async_to_lds_b128 (ASYNCcnt).

#define HMM_B 64
#define HMM_T 4096
#define HMM_N 32
#define HMM_D 64
#define TC    8              // timesteps per chunk
#define NCH   (HMM_T / TC)   // 512 chunks
#define LOG_2PI 1.8378770664093454f

typedef float v2f __attribute__((ext_vector_type(2)));
typedef float v8f __attribute__((ext_vector_type(8)));

static __device__ __forceinline__ v8f wmma4(v2f a, v2f b, v8f c) {
  // D = A(16x4) * B(4x16) + C, full f32
  return __builtin_amdgcn_wmma_f32_16x16x4_f32(false, a, false, b, (short)0, c,
                                               false, false);
}

// Generic (flat) LDS pointer -> hardware LDS byte address: low 32 bits.
static __device__ __forceinline__ unsigned lds_u32(const void* p) {
  return (unsigned)(uintptr_t)p;
}

// ---------------------------------------------------------------------------
// Prep kernel: P = row-softmax(transition), pi = softmax(priors),
// W(128x32): rows 0..63 = -0.5*ivar, rows 64..127 = mu*ivar, bias c_n.
// ws layout (floats): P @0 (1024) | pi @1024 (32) | W @1056 (4096) | bias @5152 (32)
// ---------------------------------------------------------------------------
__global__ void hmm_prep_kernel(const float* __restrict__ trans,
                                const float* __restrict__ priors,
                                const float* __restrict__ mus,
                                const float* __restrict__ lvs,
                                float* __restrict__ ws) {
  const int tid = threadIdx.x;  // blockDim = 128
  float* P    = ws;
  float* pi   = ws + 1024;
  float* W    = ws + 1056;
  float* bias = ws + 5152;

  if (tid < 32) {  // row softmax of transition row `tid`
    float m = -3.4e38f;
    for (int j = 0; j < 32; ++j) m = fmaxf(m, trans[tid * 32 + j]);
    float s = 0.f;
    for (int j = 0; j < 32; ++j) s += __expf(trans[tid * 32 + j] - m);
    float inv = 1.0f / s;
    for (int j = 0; j < 32; ++j) P[tid * 32 + j] = __expf(trans[tid * 32 + j] - m) * inv;
  } else if (tid < 64) {  // pi softmax, one state per thread
    const int n = tid - 32;
    float m = -3.4e38f;
    for (int j = 0; j < 32; ++j) m = fmaxf(m, priors[j]);
    float s = 0.f;
    for (int j = 0; j < 32; ++j) s += __expf(priors[j] - m);
    pi[n] = __expf(priors[n] - m) / s;
  } else if (tid < 96) {  // bias
    const int n = tid - 64;
    float acc = 0.f;
    for (int d = 0; d < HMM_D; ++d) {
      float mu = mus[n * HMM_D + d], lv = lvs[n * HMM_D + d];
      acc += mu * mu * __expf(-lv) + lv;
    }
    bias[n] = -0.5f * (acc + (float)HMM_D * LOG_2PI);
  }
  // W: all 128 threads strided
  for (int i = tid; i < 128 * 32; i += 128) {
    int d = i >> 5, n = i & 31;
    if (d < HMM_D) {
      W[i] = -0.5f * __expf(-lvs[n * HMM_D + d]);
    } else {
      int dd = d - HMM_D;
      W[i] = mus[n * HMM_D + dd] * __expf(-lvs[n * HMM_D + dd]);
    }
  }
}

// ---------------------------------------------------------------------------
// Async stage one tile: X[16 subj][64 dims] for one global timestep into LDS.
// 256 B128 transfers -> 8 per lane.
// ---------------------------------------------------------------------------
static __device__ __forceinline__ void stage_tile(const float* __restrict__ X,
                                                  float* xb_tile, int subj0,
                                                  int tglob, int lane) {
  const unsigned base_lds = lds_u32(xb_tile);
#pragma unroll
  for (int i = 0; i < 8; ++i) {
    int linear = i * 32 + lane;
    int subj = linear >> 4;
    int dg = linear & 15;  // group of 4 floats
    unsigned voff = (unsigned)((((subj0 + subj) * HMM_T + tglob) * HMM_D + dg * 4) * 4);
    unsigned la = base_lds + (unsigned)((subj * HMM_D + dg * 4) * 4);
    asm volatile("global_load_async_to_lds_b128 %0, %1, %2 offset:0"
                 :
                 : "v"(la), "v"(voff), "s"(X)
                 : "memory");
  }
}

// ---------------------------------------------------------------------------
// Emission tile: 16 subjects at one timestep, all 32 states.
// logB = [x^2, x] @ W + bias via 64 f32 WMMAs, then row-max + exp-shift to LDS.
// K loop is split so the squared/linear feature choice is compile-time
// (k0 < 64 <=> kk < 16, independent of the lane-dependent `hi`): no divergence.
// ---------------------------------------------------------------------------
static __device__ __forceinline__ void emit_tile(const float* W_s,
                                                 const float* bias_s,
                                                 const float* xb,     // [16][64]
                                                 float* emis_t,       // [16][32]
                                                 float* maxb_t,       // [16]
                                                 int lane) {
  const int col = lane & 15;
  const int hi = lane >> 4;
  const int la = lane & 15;  // A-matrix row (subject) held by this lane
  v8f c0, c1;
  {
    float b0v = bias_s[col], b1v = bias_s[col + 16];
#pragma unroll
    for (int r = 0; r < 8; ++r) { c0[r] = b0v; c1[r] = b1v; }
  }
  // --- K = 0..63 : feature = x^2 ---
#pragma unroll 8
  for (int kk = 0; kk < 16; ++kk) {
    const int k0 = kk * 4 + hi * 2;
    v2f a;
    a.x = xb[la * HMM_D + k0];
    a.y = xb[la * HMM_D + k0 + 1];
    a.x *= a.x;
    a.y *= a.y;
    v2f b0, b1;
    b0.x = W_s[k0 * 32 + col];        b0.y = W_s[(k0 + 1) * 32 + col];
    b1.x = W_s[k0 * 32 + col + 16];   b1.y = W_s[(k0 + 1) * 32 + col + 16];
    c0 = wmma4(a, b0, c0);
    c1 = wmma4(a, b1, c1);
  }
  // --- K = 64..127 : feature = x ---
#pragma unroll 8
  for (int kk = 16; kk < 32; ++kk) {
    const int k0 = kk * 4 + hi * 2;
    const int d = k0 - HMM_D;
    v2f a;
    a.x = xb[la * HMM_D + d];
    a.y = xb[la * HMM_D + d + 1];
    v2f b0, b1;
    b0.x = W_s[k0 * 32 + col];        b0.y = W_s[(k0 + 1) * 32 + col];
    b1.x = W_s[k0 * 32 + col + 16];   b1.y = W_s[(k0 + 1) * 32 + col + 16];
    c0 = wmma4(a, b0, c0);
    c1 = wmma4(a, b1, c1);
  }
#pragma unroll
  for (int r = 0; r < 8; ++r) {
    const int subj = r + 8 * hi;
    float v0 = c0[r], v1 = c1[r];
    float m = fmaxf(v0, v1);
    m = fmaxf(m, __shfl_xor(m, 1));
    m = fmaxf(m, __shfl_xor(m, 2));
    m = fmaxf(m, __shfl_xor(m, 4));
    m = fmaxf(m, __shfl_xor(m, 8));
    emis_t[subj * 32 + col]      = __expf(v0 - m);
    emis_t[subj * 32 + col + 16] = __expf(v1 - m);
    if (col == 0) maxb_t[subj] = m;
  }
}

// ---------------------------------------------------------------------------
// Main fused kernel: 4 blocks x 256 threads; block owns 16 subjects.
// Wave 0 = recurrence; waves 1..7 = async stage + emissions for next chunk.
// ---------------------------------------------------------------------------
__global__ __launch_bounds__(256) void hmm_fwd_kernel(const float* __restrict__ X,
                                                      const float* __restrict__ ws,
                                                      float* __restrict__ out) {
  extern __shared__ float smem[];
  float* W_s     = smem;                 // 4096 f
  float* P_s     = W_s + 4096;           // 1024 f
  float* pi_s    = P_s + 1024;           // 32 f
  float* bias_s  = pi_s + 32;            // 32 f
  float* alpha_s = bias_s + 32;          // 16*32 = 512 f
  float* maxb    = alpha_s + 512;        // 2 * TC*16 = 256 f
  float* emis    = maxb + 256;           // 2 * TC*16*32 = 8192 f
  float* xbuf    = emis + 8192;          // 2 * TC*16*64 = 16384 f
  // total = 30528 floats = 122112 bytes (within 320KB WGP LDS)

  const int tid = threadIdx.x;
  const int lane = tid & 31;
  const int wave = tid >> 5;
  const int subj0 = blockIdx.x * 16;

  // Load constants ws -> LDS
  for (int i = tid; i < 4096; i += 256) W_s[i] = ws[1056 + i];
  for (int i = tid; i < 1024; i += 256) P_s[i] = ws[i];
  if (tid < 32) { pi_s[tid] = ws[1024 + tid]; bias_s[tid] = ws[5152 + tid]; }
  __syncthreads();

  // Prologue: all 8 waves stage + emit chunk 0 (wave w -> tile w)
  {
    float* xb = xbuf + wave * (HMM_D * 16);
    stage_tile(X, xb, subj0, /*tglob=*/wave, lane);
    asm volatile("s_wait_asynccnt 0x0" ::: "memory");
    emit_tile(W_s, bias_s, xb, emis + wave * 512, maxb + wave * 16, lane);
  }
  __syncthreads();

  // Recurrence wave: hoist P B-fragments (constant over all T) into registers.
  const int col = lane & 15;
  const int hi = lane >> 4;
  const int la = lane & 15;
  v2f pf[8][2];
  float ls[8];
  if (wave == 0) {
#pragma unroll
    for (int kk = 0; kk < 8; ++kk) {
      const int k0 = kk * 4 + hi * 2;
      pf[kk][0].x = P_s[k0 * 32 + col];       pf[kk][0].y = P_s[(k0 + 1) * 32 + col];
      pf[kk][1].x = P_s[k0 * 32 + col + 16];  pf[kk][1].y = P_s[(k0 + 1) * 32 + col + 16];
    }
#pragma unroll
    for (int r = 0; r < 8; ++r) ls[r] = 0.f;
  }

  for (int k = 0; k < NCH; ++k) {
    const int cur = k & 1, nxt = cur ^ 1;
    if (wave == 0) {
      // ---- recurrence over TC steps of chunk k ----
      const float* emis_c = emis + cur * 4096;
      const float* maxb_c = maxb + cur * 128;
      for (int tt = 0; tt < TC; ++tt) {
        const int t = k * TC + tt;
        v8f c0 = {}, c1 = {};
        if (t == 0) {
          const float p0 = pi_s[col], p1 = pi_s[col + 16];
#pragma unroll
          for (int r = 0; r < 8; ++r) { c0[r] = p0; c1[r] = p1; }
        } else {
#pragma unroll
          for (int kk = 0; kk < 8; ++kk) {
            const int k0 = kk * 4 + hi * 2;
            v2f a;
            a.x = alpha_s[la * 32 + k0];
            a.y = alpha_s[la * 32 + k0 + 1];
            c0 = wmma4(a, pf[kk][0], c0);
            c1 = wmma4(a, pf[kk][1], c1);
          }
        }
        const float* et = emis_c + tt * 512;
        const float* mt = maxb_c + tt * 16;
#pragma unroll
        for (int r = 0; r < 8; ++r) {
          const int subj = r + 8 * hi;
          float v0 = c0[r] * et[subj * 32 + col];
          float v1 = c1[r] * et[subj * 32 + col + 16];
          float s = v0 + v1;
          s += __shfl_xor(s, 1);
          s += __shfl_xor(s, 2);
          s += __shfl_xor(s, 4);
          s += __shfl_xor(s, 8);
          float inv = 1.0f / s;
          alpha_s[subj * 32 + col]      = v0 * inv;
          alpha_s[subj * 32 + col + 16] = v1 * inv;
          ls[r] += mt[subj] + __logf(s);
          if (col == 0) out[(subj0 + subj) * HMM_T + t] = ls[r];
        }
      }
    } else if (k + 1 < NCH) {
      // ---- waves 1..7: async stage + emissions for chunk k+1 ----
      for (int tt = wave - 1; tt < TC; tt += 7) {
        stage_tile(X, xbuf + nxt * 8192 + tt * 1024, subj0, (k + 1) * TC + tt, lane);
      }
      asm volatile("s_wait_asynccnt 0x0" ::: "memory");
      for (int tt = wave - 1; tt < TC; tt += 7) {
        emit_tile(W_s, bias_s, xbuf + nxt * 8192 + tt * 1024,
                  emis + nxt * 4096 + tt * 512, maxb + nxt * 128 + tt * 16, lane);
      }
    }
    __syncthreads();
  }
}

// ---------------------------------------------------------------------------
extern "C" void kernel_launch(void* const* d_in, const int* in_sizes, int n_in,
                              void* d_out, int out_size, void* d_ws, size_t ws_size,
                              hipStream_t stream) {
  const float* X      = (const float*)d_in[0];  // (B,T,D)
  const float* trans  = (const float*)d_in[1];  // (N,N)
  const float* priors = (const float*)d_in[2];  // (N,)
  const float* mus    = (const float*)d_in[3];  // (N,D)
  const float* lvs    = (const float*)d_in[4];  // (N,D)
  float* ws  = (float*)d_ws;
  float* out = (float*)d_out;                   // (B,T)

  hipLaunchKernelGGL(hmm_prep_kernel, dim3(1), dim3(128), 0, stream,
                     trans, priors, mus, lvs, ws);

  const size_t shmem = 30528u * sizeof(float);  // ~119 KB dynamic LDS
  hipLaunchKernelGGL(hmm_fwd_kernel, dim3(HMM_B / 16), dim3(256), shmem, stream,
                     X, ws, out);
}